// GATRegressor_67765993997188
// MI455X (gfx1250) — compile-verified
//
#include <hip/hip_runtime.h>
#include <math.h>

typedef float v2f __attribute__((ext_vector_type(2)));
typedef float v8f __attribute__((ext_vector_type(8)));

#define N_NODES 50000
#define N_EDGES 800000
#define IN_DIM  160
#define HID     64
#define HEADS   4
#define LAYERS  3
#define N_GRAPHS 128
#define E_TOT   (N_EDGES + N_NODES)   // edges + self loops
#define MAX_K   160
#define LDS_PAD 4

// ---------------------------------------------------------------- utilities

__global__ void fill_kernel(float* __restrict__ p, float v, long n) {
    long i = (long)blockIdx.x * blockDim.x + threadIdx.x;
    if (i < n) p[i] = v;
}

__device__ __forceinline__ void edge_nodes(const int* __restrict__ ei, int e,
                                           int& s, int& d) {
    if (e < N_EDGES) { s = ei[e]; d = ei[N_EDGES + e]; }
    else             { s = d = e - N_EDGES; }
}

__device__ __forceinline__ float leaky(float v) { return v > 0.f ? v : 0.2f * v; }

// float atomic max via ordered-int trick (init must be -inf)
__device__ __forceinline__ void atomicMaxF(float* addr, float val) {
    if (val >= 0.f) atomicMax((int*)addr, __float_as_int(val));
    else            atomicMin((unsigned int*)addr, __float_as_uint(val));
}

// CDNA5 async global->LDS copy (16B per lane, ASYNCcnt-tracked).
// lds_off is a byte offset from the workgroup LDS base.
__device__ __forceinline__ void async_g2l_b128(const float* gsrc, unsigned lds_off) {
    asm volatile("global_load_async_to_lds_b128 %0, %1, off"
                 :: "v"(lds_off), "v"(gsrc) : "memory");
}
__device__ __forceinline__ void wait_async0() {
    asm volatile("s_wait_asynccnt 0x0" ::: "memory");
}

// ------------------------------------------------- WMMA fp32 GEMM (16x16x4)
// C[M,N] = A[M,K] @ B[K,N] (+bias) (+relu). Row-major. M%16==0, N%16==0,
// K%4==0, K<=160. One block = 16 rows; the 16xK A tile is staged into LDS
// once via async-to-LDS, then all 4 waves (one 16x16 tile each, 64 cols per
// block slice) consume it. B (tiny, L2-resident) is read directly.
__global__ __launch_bounds__(128)
void wmma_gemm_f32(const float* __restrict__ A, const float* __restrict__ B,
                   const float* __restrict__ bias, float* __restrict__ C,
                   int M, int N, int K, int act)
{
    __shared__ float sA[16 * (MAX_K + LDS_PAD)];    // LDS offset 0 (only alloc)
    const int tid  = threadIdx.x;
    const int wave = tid >> 5;
    const int lane = tid & 31;
    const int m0   = blockIdx.x * 16;
    const int lds_stride = K + LDS_PAD;

    // ---- stage A[m0:m0+16, 0:K] into LDS, coalesced 16B async copies
    const int kvecs = K >> 2;           // b128 chunks per row
    const int tasks = 16 * kvecs;
    for (int i = tid; i < tasks; i += 128) {
        int row = i / kvecs;
        int kk  = (i - row * kvecs) << 2;
        async_g2l_b128(A + (long)(m0 + row) * K + kk,
                       (unsigned)((row * lds_stride + kk) * 4));
    }
    wait_async0();
    __syncthreads();

    const int n0 = (blockIdx.y * 4 + wave) * 16;
    if (n0 < N) {                        // uniform per wave: EXEC stays all-1s
        const int col   = n0 + (lane & 15);
        const int arow  = (lane & 15) * lds_stride;
        const int koff  = (lane >> 4) << 1;   // lanes 0-15: K+0/1, 16-31: K+2/3
        v8f acc = {};
        for (int k0 = 0; k0 < K; k0 += 4) {
            v2f a;                       // A fragment from LDS
            a.x = sA[arow + k0 + koff];
            a.y = sA[arow + k0 + koff + 1];
            v2f b;                       // B fragment from global (L2-resident)
            b.x = B[(long)(k0 + koff    ) * N + col];
            b.y = B[(long)(k0 + koff + 1) * N + col];
            acc = __builtin_amdgcn_wmma_f32_16x16x4_f32(
                      false, a, false, b, (short)0, acc, false, false);
        }
        // C layout: VGPR r -> M = r (lanes 0-15) / r+8 (lanes 16-31)
        const int mbase = m0 + ((lane >> 4) << 3);
        const float bi = bias ? bias[col] : 0.f;
#pragma unroll
        for (int r = 0; r < 8; ++r) {
            float v = acc[r] + bi;
            if (act == 1) v = v > 0.f ? v : 0.f;
            C[(long)(mbase + r) * N + col] = v;
        }
    }
}

// ------------------------------------------------------- attention scores
// a_s[n,h] = dot(xh[n,h,:], att_src[h,:]) ; a_d likewise.
__global__ void att_scores(const float* __restrict__ xh,
                           const float* __restrict__ att_s,
                           const float* __restrict__ att_d,
                           float* __restrict__ a_s, float* __restrict__ a_d)
{
    int t = blockIdx.x * blockDim.x + threadIdx.x;
    if (t >= N_NODES * HEADS) return;
    int n = t >> 2, h = t & 3;
    const float* xp = xh + (long)n * (HEADS * HID) + h * HID;
    const float* ws = att_s + h * HID;
    const float* wd = att_d + h * HID;
    float ss = 0.f, sd = 0.f;
#pragma unroll 8
    for (int c = 0; c < HID; ++c) { float v = xp[c]; ss += v * ws[c]; sd += v * wd[c]; }
    a_s[t] = ss; a_d[t] = sd;
}

// --------------------------------------------------- segment softmax passes
__global__ void edge_max(const int* __restrict__ ei,
                         const float* __restrict__ a_s, const float* __restrict__ a_d,
                         float* __restrict__ emax)
{
    int t = blockIdx.x * blockDim.x + threadIdx.x;
    if (t >= E_TOT * HEADS) return;
    int e = t >> 2, h = t & 3, s, d;
    edge_nodes(ei, e, s, d);
    float v = leaky(a_s[s * 4 + h] + a_d[d * 4 + h]);
    atomicMaxF(&emax[d * 4 + h], v);
}

__global__ void edge_expsum(const int* __restrict__ ei,
                            const float* __restrict__ a_s, const float* __restrict__ a_d,
                            const float* __restrict__ emax,
                            float* __restrict__ ee, float* __restrict__ denom)
{
    int t = blockIdx.x * blockDim.x + threadIdx.x;
    if (t >= E_TOT * HEADS) return;
    int e = t >> 2, h = t & 3, s, d;
    edge_nodes(ei, e, s, d);
    float v = leaky(a_s[s * 4 + h] + a_d[d * 4 + h]);
    float x = __expf(v - emax[d * 4 + h]);
    ee[t] = x;
    atomicAdd(&denom[d * 4 + h], x);
}

__global__ void edge_alpha(const int* __restrict__ ei,
                           const float* __restrict__ denom, float* __restrict__ ee)
{
    int t = blockIdx.x * blockDim.x + threadIdx.x;
    if (t >= E_TOT * HEADS) return;
    int e = t >> 2, h = t & 3, s, d;
    edge_nodes(ei, e, s, d);
    ee[t] = ee[t] / denom[d * 4 + h];
}

// ------------------------------- message scatter: coalesced 256 lanes/edge
__global__ __launch_bounds__(256)
void edge_message(const int* __restrict__ ei, const float* __restrict__ alpha,
                  const float* __restrict__ xh, float* __restrict__ agg)
{
    int e  = blockIdx.x;            // one block per edge (uniform -> s_load)
    int ch = threadIdx.x;           // channel 0..255, head = ch>>6
    int h  = ch >> 6, s, d;
    edge_nodes(ei, e, s, d);
    float a = alpha[e * 4 + h];
    atomicAdd(&agg[(long)d * (HEADS * HID) + ch],
              a * xh[(long)s * (HEADS * HID) + ch]);
}

// ----------------------------------------- head-mean + bias + ELU + residual
__global__ void node_update(const float* __restrict__ agg,
                            const float* __restrict__ gb, float* __restrict__ h)
{
    int t = blockIdx.x * blockDim.x + threadIdx.x;
    if (t >= N_NODES * HID) return;
    int n = t >> 6, c = t & 63;
    const float* ap = agg + (long)n * (HEADS * HID) + c;
    float m = 0.25f * (ap[0] + ap[64] + ap[128] + ap[192]) + gb[c];
    float el = m > 0.f ? m : expm1f(m);      // ELU(alpha=1)
    h[t] = el + h[t];
}

// --------------------------------------------------------------- pooling
__global__ void pool_sum(const float* __restrict__ h, const int* __restrict__ batch,
                         float* __restrict__ sums)
{
    int t = blockIdx.x * blockDim.x + threadIdx.x;
    if (t >= N_NODES * HID) return;
    int n = t >> 6, c = t & 63;
    atomicAdd(&sums[batch[n] * HID + c], h[t]);
}

__global__ void pool_count(const int* __restrict__ batch, float* __restrict__ counts)
{
    int n = blockIdx.x * blockDim.x + threadIdx.x;
    if (n >= N_NODES) return;
    atomicAdd(&counts[batch[n]], 1.f);
}

__global__ void pool_div(const float* __restrict__ sums, const float* __restrict__ counts,
                         float* __restrict__ pooled)
{
    int t = blockIdx.x * blockDim.x + threadIdx.x;
    if (t >= N_GRAPHS * HID) return;
    int g = t >> 6;
    pooled[t] = sums[t] / fmaxf(counts[g], 1.f);
}

// ---------------------------------------------------------- final fc2 matvec
__global__ __launch_bounds__(64)
void fc2_kernel(const float* __restrict__ z, const float* __restrict__ w,
                const float* __restrict__ b, float* __restrict__ out)
{
    __shared__ float red[64];
    int g = blockIdx.x, c = threadIdx.x;
    red[c] = z[g * HID + c] * w[c];
    __syncthreads();
    for (int s = 32; s > 0; s >>= 1) {
        if (c < s) red[c] += red[c + s];
        __syncthreads();
    }
    if (c == 0) out[g] = red[0] + b[0];
}

// ================================================================= launcher
extern "C" void kernel_launch(void* const* d_in, const int* in_sizes, int n_in,
                              void* d_out, int out_size, void* d_ws, size_t ws_size,
                              hipStream_t stream) {
    (void)in_sizes; (void)n_in; (void)out_size; (void)ws_size;
    const float* x       = (const float*)d_in[0];
    const int*   ei      = (const int*)  d_in[1];
    const int*   batch   = (const int*)  d_in[2];
    /* d_in[3] = num_graphs (==128, compile-time constant) */
    const float* in_w    = (const float*)d_in[4];
    const float* in_b    = (const float*)d_in[5];
    const float* gat_w   = (const float*)d_in[6];
    const float* att_src = (const float*)d_in[7];
    const float* att_dst = (const float*)d_in[8];
    const float* gat_b   = (const float*)d_in[9];
    const float* fc1_w   = (const float*)d_in[10];
    const float* fc1_b   = (const float*)d_in[11];
    const float* fc2_w   = (const float*)d_in[12];
    const float* fc2_b   = (const float*)d_in[13];
    float* out = (float*)d_out;

    // -------- workspace carve-up (floats)
    float* ws = (float*)d_ws;
    size_t off = 0;
    float* h      = ws + off; off += (size_t)N_NODES * HID;           // 3.2M
    float* xh     = ws + off; off += (size_t)N_NODES * HEADS * HID;   // 12.8M
    float* a_s    = ws + off; off += (size_t)N_NODES * HEADS;
    float* a_d    = ws + off; off += (size_t)N_NODES * HEADS;
    float* emax   = ws + off; off += (size_t)N_NODES * HEADS;
    float* denom  = ws + off; off += (size_t)N_NODES * HEADS;
    float* ee     = ws + off; off += (size_t)E_TOT * HEADS;           // 3.4M
    float* agg    = ws + off; off += (size_t)N_NODES * HEADS * HID;   // 12.8M
    float* sums   = ws + off; off += (size_t)N_GRAPHS * HID;
    float* counts = ws + off; off += (size_t)N_GRAPHS;
    float* pooled = ws + off; off += (size_t)N_GRAPHS * HID;
    float* z      = ws + off; off += (size_t)N_GRAPHS * HID;

    const int T = 256;
    auto blocks = [](long n, int t) { return (unsigned)((n + t - 1) / t); };

    // -------- input projection: h = x @ in_w + in_b   (WMMA fp32)
    wmma_gemm_f32<<<dim3(N_NODES / 16, HID / 64), 128, 0, stream>>>(
        x, in_w, in_b, h, N_NODES, HID, IN_DIM, 0);

    // -------- GAT layers
    for (int l = 0; l < LAYERS; ++l) {
        const float* gw = gat_w   + (size_t)l * HID * HEADS * HID;
        const float* as = att_src + (size_t)l * HEADS * HID;
        const float* ad = att_dst + (size_t)l * HEADS * HID;
        const float* gb = gat_b   + (size_t)l * HID;

        // xh = h @ gat_w[l]   (50000 x 64 x 256, WMMA fp32)
        wmma_gemm_f32<<<dim3(N_NODES / 16, (HEADS * HID) / 64), 128, 0, stream>>>(
            h, gw, nullptr, xh, N_NODES, HEADS * HID, HID, 0);

        att_scores<<<blocks((long)N_NODES * HEADS, T), T, 0, stream>>>(xh, as, ad, a_s, a_d);

        fill_kernel<<<blocks((long)N_NODES * HEADS, T), T, 0, stream>>>(
            emax, -__builtin_huge_valf(), (long)N_NODES * HEADS);
        fill_kernel<<<blocks((long)N_NODES * HEADS, T), T, 0, stream>>>(
            denom, 0.f, (long)N_NODES * HEADS);
        fill_kernel<<<blocks((long)N_NODES * HEADS * HID, T), T, 0, stream>>>(
            agg, 0.f, (long)N_NODES * HEADS * HID);

        edge_max   <<<blocks((long)E_TOT * HEADS, T), T, 0, stream>>>(ei, a_s, a_d, emax);
        edge_expsum<<<blocks((long)E_TOT * HEADS, T), T, 0, stream>>>(ei, a_s, a_d, emax, ee, denom);
        edge_alpha <<<blocks((long)E_TOT * HEADS, T), T, 0, stream>>>(ei, denom, ee);
        edge_message<<<E_TOT, 256, 0, stream>>>(ei, ee, xh, agg);

        node_update<<<blocks((long)N_NODES * HID, T), T, 0, stream>>>(agg, gb, h);
    }

    // -------- global mean pool + MLP head
    fill_kernel<<<blocks((long)N_GRAPHS * (HID + 1), T), T, 0, stream>>>(
        sums, 0.f, (long)N_GRAPHS * (HID + 1));   // sums + counts are contiguous
    pool_sum  <<<blocks((long)N_NODES * HID, T), T, 0, stream>>>(h, batch, sums);
    pool_count<<<blocks(N_NODES, T), T, 0, stream>>>(batch, counts);
    pool_div  <<<blocks((long)N_GRAPHS * HID, T), T, 0, stream>>>(sums, counts, pooled);

    // z = relu(pooled @ fc1_w + fc1_b)  (128 x 64 x 64, WMMA fp32)
    wmma_gemm_f32<<<dim3(N_GRAPHS / 16, 1), 128, 0, stream>>>(
        pooled, fc1_w, fc1_b, z, N_GRAPHS, HID, HID, 1);

    fc2_kernel<<<N_GRAPHS, 64, 0, stream>>>(z, fc2_w, fc2_b, out);
}